// AdditiveRecursiveNN_23733989278139
// MI455X (gfx1250) — compile-verified
//
#include <hip/hip_runtime.h>
#include <hip/hip_bf16.h>

typedef __attribute__((ext_vector_type(2))) float v2f;
typedef __attribute__((ext_vector_type(4))) float v4f;
typedef __attribute__((ext_vector_type(8))) float v8f;

#define EDIM   32
#define LROW   36      // padded LDS row stride (floats): bank-conflict-free, 16B aligned
#define WAVES  4       // 4 waves (nodes) per block -> 128 threads, 36 KB LDS

__device__ __forceinline__ v4f relu4(v4f v) {
    v4f r;
    r.x = fmaxf(v.x, 0.0f); r.y = fmaxf(v.y, 0.0f);
    r.z = fmaxf(v.z, 0.0f); r.w = fmaxf(v.w, 0.0f);
    return r;
}

// One wave computes one internal node: inp = childL + childR (relu'd leaves at
// the first level), out = relu(inp @ W^T + b).  32x32x32 fp32 matmul done as
// 2x2 tiles of V_WMMA_F32_16X16X4_F32, K accumulated over 8 chunks of 4.
__global__ __launch_bounds__(WAVES * 32)
void tree_level_wmma(const float* __restrict__ emb,
                     const float* __restrict__ bias_tbl,
                     const int*   __restrict__ leaf_ids,
                     const int*   __restrict__ internal_ids,
                     const float* __restrict__ child,
                     float*       __restrict__ out,
                     int n_nodes, int id_offset, int leaf_level)
{
    __shared__ float lds[WAVES][2][EDIM * LROW];   // [wave][0]=inp, [1]=W
    const int lane = threadIdx.x & 31;
    const int wave = threadIdx.x >> 5;
    int node = blockIdx.x * WAVES + wave;
    if (node >= n_nodes) node = n_nodes - 1;       // clamp: keep barriers uniform

    float* inpL = &lds[wave][0][0];
    float* wL   = &lds[wave][1][0];

    const int    wid = internal_ids[id_offset + node];
    const float* Wg  = emb + (size_t)wid * (EDIM * EDIM);

    const float* c0;
    const float* c1;
    if (leaf_level) {
        c0 = emb + (size_t)leaf_ids[2 * node]     * (EDIM * EDIM);
        c1 = emb + (size_t)leaf_ids[2 * node + 1] * (EDIM * EDIM);
    } else {
        c0 = child + (size_t)(2 * node)     * (EDIM * EDIM);
        c1 = child + (size_t)(2 * node + 1) * (EDIM * EDIM);
    }

    // Cooperative fill: 8 x float4 per lane per matrix (coalesced global_load_b128)
    #pragma unroll
    for (int r = 0; r < 8; ++r) {
        const int gidx = r * 128 + lane * 4;       // element index in 32x32 row-major
        const int grow = gidx >> 5;
        const int gcol = gidx & 31;
        v4f a = *(const v4f*)(c0 + gidx);
        v4f b = *(const v4f*)(c1 + gidx);
        if (leaf_level) { a = relu4(a); b = relu4(b); }
        v4f s;
        s.x = a.x + b.x; s.y = a.y + b.y; s.z = a.z + b.z; s.w = a.w + b.w;
        *(v4f*)(inpL + grow * LROW + gcol) = s;
        *(v4f*)(wL   + grow * LROW + gcol) = *(const v4f*)(Wg + gidx);
    }
    __syncthreads();

    const int    half = lane >> 4;                 // K-half select (ISA A/B layout)
    const int    l16  = lane & 15;
    const float* brow = bias_tbl + (size_t)wid * EDIM;
    float*       outp = out + (size_t)node * (EDIM * EDIM);

    #pragma unroll
    for (int tm = 0; tm < 2; ++tm) {
        #pragma unroll
        for (int tn = 0; tn < 2; ++tn) {
            v8f acc = {};
            const int arow = tm * 16 + l16;        // A: M row
            const int ncol = tn * 16 + l16;        // B/C: N col (== output dim k)
            #pragma unroll
            for (int kc = 0; kc < 8; ++kc) {
                const int k0 = kc * 4 + half * 2;
                v2f a = *(const v2f*)(inpL + arow * LROW + k0);       // A[M][k0..k0+1]
                v2f b = *(const v2f*)(wL   + ncol * LROW + k0);       // B[j][N] = W[N][j]
                acc = __builtin_amdgcn_wmma_f32_16x16x4_f32(
                        false, a, false, b, (short)0, acc, false, false);
            }
            const float bv = brow[ncol];           // bias broadcast over rows
            #pragma unroll
            for (int v = 0; v < 8; ++v) {
                const int mrow = tm * 16 + half * 8 + v;   // C/D VGPR layout
                outp[mrow * EDIM + ncol] = fmaxf(acc[v] + bv, 0.0f);
            }
        }
    }
}

// Head: logits = flat(cur) @ proj_W^T + proj_b ; log-softmax ; loss ; argmax.
__global__ __launch_bounds__(256)
void tree_head(const float* __restrict__ cur,
               const float* __restrict__ proj_W,
               const float* __restrict__ proj_b,
               const int*   __restrict__ label,
               float*       __restrict__ out)
{
    __shared__ float s0[256];
    __shared__ float s1[256];
    const int tid = threadIdx.x;
    float p0 = 0.0f, p1 = 0.0f;
    for (int i = tid; i < EDIM * EDIM; i += 256) {
        const float x = cur[i];
        p0 = fmaf(x, proj_W[i], p0);
        p1 = fmaf(x, proj_W[EDIM * EDIM + i], p1);
    }
    s0[tid] = p0; s1[tid] = p1;
    __syncthreads();
    for (int s = 128; s > 0; s >>= 1) {
        if (tid < s) { s0[tid] += s0[tid + s]; s1[tid] += s1[tid + s]; }
        __syncthreads();
    }
    if (tid == 0) {
        const float l0 = s0[0] + proj_b[0];
        const float l1 = s1[0] + proj_b[1];
        const int   pred = (l1 > l0) ? 1 : 0;       // jnp.argmax: first max wins ties
        const float m    = fmaxf(l0, l1);
        const float lse  = m + logf(expf(l0 - m) + expf(l1 - m));
        const int   lab  = label[0];
        const float logp = ((lab == 0) ? l0 : l1) - lse;
        out[0] = (float)pred;
        out[1] = -logp;
    }
}

extern "C" void kernel_launch(void* const* d_in, const int* in_sizes, int n_in,
                              void* d_out, int out_size, void* d_ws, size_t ws_size,
                              hipStream_t stream) {
    (void)in_sizes; (void)n_in; (void)out_size; (void)ws_size;
    const float* emb          = (const float*)d_in[0];   // (VOCAB, 1024)
    const float* bias_tbl     = (const float*)d_in[1];   // (VOCAB, 32)
    const float* proj_W       = (const float*)d_in[2];   // (2, 1024)
    const float* proj_b       = (const float*)d_in[3];   // (2,)
    const int*   leaf_ids     = (const int*)d_in[4];     // (4096,)
    const int*   internal_ids = (const int*)d_in[5];     // (4095,)
    const int*   label        = (const int*)d_in[6];     // scalar
    float*       out          = (float*)d_out;           // [pred, loss]

    float* bufA = (float*)d_ws;                          // 2048 * 1024 floats (8 MB)
    float* bufB = bufA + (size_t)2048 * 1024;            // 1024 * 1024 floats (4 MB)

    // Level 1: 2048 nodes from relu'd leaf pairs.
    int n = 2048;
    tree_level_wmma<<<(n + WAVES - 1) / WAVES, WAVES * 32, 0, stream>>>(
        emb, bias_tbl, leaf_ids, internal_ids, nullptr, bufA, n, 0, 1);
    int    offset = n;
    float* cur    = bufA;
    float* next   = bufB;

    // Levels 2..12: 1024 -> 1 nodes, ping-pong in workspace.
    for (n = 1024; n >= 1; n >>= 1) {
        tree_level_wmma<<<(n + WAVES - 1) / WAVES, WAVES * 32, 0, stream>>>(
            emb, bias_tbl, leaf_ids, internal_ids, cur, next, n, offset, 0);
        offset += n;
        float* t = cur; cur = next; next = t;
    }

    tree_head<<<1, 256, 0, stream>>>(cur, proj_W, proj_b, label, out);
}